// FFJORD_56573309223751
// MI455X (gfx1250) — compile-verified
//
#include <hip/hip_runtime.h>

typedef float v2f __attribute__((ext_vector_type(2)));
typedef float v8f __attribute__((ext_vector_type(8)));

#define NSTEPS   16
#define FD_EPS_F 1e-4f
#define BTOT     65536
#define DD       64
#define HH       256
#define NW       4            // waves per block (wave32)
#define ROWS_PW  16           // batch rows per wave
#define BT       (NW * ROWS_PW)
#define XSTR     72           // 64 x-cols + t col + 3 zero-pad (K padded to 68)
#define HSTR     264          // 256 + pad
#define K0PAD    68           // layer-0 K, zero padded

// ---------------------------------------------------------------------------
// f(t, x) = x + MLP([x | t]) for a 16-row slab held in LDS.
// X (16 x XSTR) holds [x | t | 0 0 0] on entry and f(t,x) in cols 0..63 on exit.
// All GEMMs use V_WMMA_F32_16X16X4_F32 (fp32 — the FD trace with eps=1e-4
// destroys sub-fp32 data paths, and the workload is compute-bound:
// ~1 TFLOP vs ~300 MB of HBM traffic at 23.3 TB/s).
// W0/W2/biases come from LDS (staged once per block, reused 80 evals);
// W1 (256 KB) streams from the 192 MB L2 where it is permanently resident.
// WMMA f32 16x16x4 fragment layout (wave32):
//   A (16x4): lane<16 -> row=lane, {K=0,K=1}; lane>=16 -> row=lane-16, {K=2,K=3}
//   B (4x16): lane<16 -> col=lane, {K=0,K=1}; lane>=16 -> col=lane-16, {K=2,K=3}
//   C (16x16): vgpr j on lane L -> M = j + 8*(L>=16), N = L%16
// ---------------------------------------------------------------------------
__device__ __forceinline__ void f_eval(float (*X)[XSTR], float (*h0)[HSTR], float (*h1)[HSTR],
                                       const float (*W0s)[HH], const float* b0s,
                                       const float* __restrict__ W1, const float* b1s,
                                       const float (*W2s)[DD], const float* b2s,
                                       int lane)
{
  const int row   = lane & 15;          // A-fragment M index
  const int khalf = (lane >> 4) << 1;   // A/B-fragment K sub-offset (0 or 2)
  const int ncol  = lane & 15;          // B/C-fragment N index
  const int mhi   = (lane >> 4) << 3;   // C-fragment M offset (0 or 8)

  __builtin_amdgcn_wave_barrier();

  // ----- layer 0: h0 = relu([x|t] @ W0 + b0), K = 65 zero-padded to 68 (all-LDS)
  for (int nt = 0; nt < 16; ++nt) {
    const int n  = nt * 16 + ncol;
    const float bv = b0s[n];
    v8f c;
    #pragma unroll
    for (int j = 0; j < 8; ++j) c[j] = bv;
    #pragma unroll 4
    for (int kt = 0; kt < 17; ++kt) {
      const int kk = kt * 4 + khalf;
      v2f a, b;
      a.x = X[row][kk];
      a.y = X[row][kk + 1];
      b.x = W0s[kk][n];              // zero rows 65..67 already staged
      b.y = W0s[kk + 1][n];
      c = __builtin_amdgcn_wmma_f32_16x16x4_f32(false, a, false, b, (short)0, c, false, false);
    }
    #pragma unroll
    for (int j = 0; j < 8; ++j) h0[mhi + j][n] = fmaxf(c[j], 0.f);
  }
  __builtin_amdgcn_wave_barrier();

  // ----- layer 1: h1 = relu(h0 @ W1 + b1), K = 256 (A from LDS, B from L2)
  for (int nt = 0; nt < 16; ++nt) {
    const int n  = nt * 16 + ncol;
    const float bv = b1s[n];
    v8f c;
    #pragma unroll
    for (int j = 0; j < 8; ++j) c[j] = bv;
    #pragma unroll 4
    for (int kt = 0; kt < 64; ++kt) {
      const int kk = kt * 4 + khalf;
      v2f a, b;
      a.x = h0[row][kk];
      a.y = h0[row][kk + 1];
      b.x = W1[(size_t)kk * HH + n];
      b.y = W1[(size_t)(kk + 1) * HH + n];
      c = __builtin_amdgcn_wmma_f32_16x16x4_f32(false, a, false, b, (short)0, c, false, false);
    }
    #pragma unroll
    for (int j = 0; j < 8; ++j) h1[mhi + j][n] = fmaxf(c[j], 0.f);
  }
  __builtin_amdgcn_wave_barrier();

  // ----- layer 2 (residual, in place): X[:, :64] = X[:, :64] + h1 @ W2 + b2 (all-LDS)
  for (int nt = 0; nt < 4; ++nt) {
    const int n  = nt * 16 + ncol;
    const float bv = b2s[n];
    v8f c;
    #pragma unroll
    for (int j = 0; j < 8; ++j) c[j] = bv + X[mhi + j][n];   // residual folded into C init
    #pragma unroll 4
    for (int kt = 0; kt < 64; ++kt) {
      const int kk = kt * 4 + khalf;
      v2f a, b;
      a.x = h1[row][kk];
      a.y = h1[row][kk + 1];
      b.x = W2s[kk][n];
      b.y = W2s[kk + 1][n];
      c = __builtin_amdgcn_wmma_f32_16x16x4_f32(false, a, false, b, (short)0, c, false, false);
    }
    #pragma unroll
    for (int j = 0; j < 8; ++j) X[mhi + j][n] = c[j];
  }
  __builtin_amdgcn_wave_barrier();
}

// slab <-> register helpers: lane holds flat elements lane + 32*i of the 16x64 slab
__device__ __forceinline__ void slab_store(float (*X)[XSTR], const float* v, float t, int lane) {
  #pragma unroll
  for (int i = 0; i < 32; ++i) { const int f = lane + 32 * i; X[f >> 6][f & 63] = v[i]; }
  if (lane < 16) { X[lane][64] = t; X[lane][65] = 0.f; X[lane][66] = 0.f; X[lane][67] = 0.f; }
  __builtin_amdgcn_wave_barrier();
}
__device__ __forceinline__ void slab_load(const float (*X)[XSTR], float* v, int lane) {
  __builtin_amdgcn_wave_barrier();
  #pragma unroll
  for (int i = 0; i < 32; ++i) { const int f = lane + 32 * i; v[i] = X[f >> 6][f & 63]; }
}

__launch_bounds__(NW * 32)
__global__ void ffjord_rk4_wmma(const float* __restrict__ x,
                                const float* __restrict__ eps,
                                const float* __restrict__ W0, const float* __restrict__ b0,
                                const float* __restrict__ W1, const float* __restrict__ b1,
                                const float* __restrict__ W2, const float* __restrict__ b2,
                                float* __restrict__ out)
{
  // per-wave activation slabs (wave-private)
  __shared__ float Xs[NW][ROWS_PW][XSTR];
  __shared__ float H0[NW][ROWS_PW][HSTR];
  __shared__ float H1[NW][ROWS_PW][HSTR];
  // block-shared weights, staged once and reused for all 16 steps x 5 evals
  __shared__ float W0s[K0PAD][HH];      // rows 65..67 zero (K padding incl. t row 64)
  __shared__ float W2s[HH][DD];
  __shared__ float b0s[HH], b1s[HH], b2s[DD];

  const int tid  = threadIdx.x;
  const int lane = tid & 31;
  const int wave = tid >> 5;
  const int rowbase = blockIdx.x * BT + wave * ROWS_PW;
  const size_t slab = (size_t)rowbase * DD;

  // ---- cooperative one-time weight staging into LDS
  for (int i = tid; i < K0PAD * HH; i += NW * 32) {
    const int r = i >> 8;
    (&W0s[0][0])[i] = (r < DD + 1) ? W0[i] : 0.f;
  }
  for (int i = tid; i < HH * DD; i += NW * 32) (&W2s[0][0])[i] = W2[i];
  for (int i = tid; i < HH; i += NW * 32) { b0s[i] = b0[i]; b1s[i] = b1[i]; }
  if (tid < DD) b2s[tid] = b2[tid];
  // W1 stays in L2 (192 MB, weights total 0.4 MB) — pull it in early
  __builtin_prefetch(W1, 0, 3);
  __syncthreads();

  float (*X)[XSTR]  = Xs[wave];
  float (*h0)[HSTR] = H0[wave];
  float (*h1)[HSTR] = H1[wave];

  float xc[32];
  #pragma unroll
  for (int i = 0; i < 32; ++i) xc[i] = x[slab + lane + 32 * i];

  float ld = 0.f;                      // log-det, valid on lanes 0..15 (row = rowbase+lane)
  const float dt = 1.0f / NSTEPS;

  for (int step = 0; step < NSTEPS; ++step) {
    const float t = step * dt;
    const float* es = eps + (size_t)step * BTOT * DD + slab;
    __builtin_prefetch(es, 0, 1);

    // ---- base = f(t, xc)  (doubles as RK4 k1)
    slab_store(X, xc, t, lane);
    f_eval(X, h0, h1, W0s, b0s, W1, b1s, W2s, b2s, lane);
    float base[32];
    slab_load(X, base, lane);

    // ---- plus = f(t, xc + 0.5*eps*FD_EPS)  -> Hutchinson FD trace
    float kv[32];
    #pragma unroll
    for (int i = 0; i < 32; ++i) kv[i] = xc[i] + 0.5f * FD_EPS_F * es[lane + 32 * i];
    slab_store(X, kv, t, lane);
    f_eval(X, h0, h1, W0s, b0s, W1, b1s, W2s, b2s, lane);
    // per-element e*(plus - base) into scratch (h0 is free now), then per-row sum
    __builtin_amdgcn_wave_barrier();
    #pragma unroll
    for (int i = 0; i < 32; ++i) {
      const int f = lane + 32 * i;
      h0[f >> 6][f & 63] = es[f] * (X[f >> 6][f & 63] - base[i]);
    }
    __builtin_amdgcn_wave_barrier();
    if (lane < 16) {
      float s = 0.f;
      #pragma unroll 8
      for (int c = 0; c < 64; ++c) s += h0[lane][c];
      ld -= (s / FD_EPS_F) * dt;       // trace * dt
    }

    // ---- RK4:  k1 = base
    float acc[32];
    #pragma unroll
    for (int i = 0; i < 32; ++i) acc[i] = base[i];

    // k2 = f(t + dt/2, xc + dt/2 * k1)
    #pragma unroll
    for (int i = 0; i < 32; ++i) kv[i] = xc[i] + 0.5f * dt * base[i];
    slab_store(X, kv, t + 0.5f * dt, lane);
    f_eval(X, h0, h1, W0s, b0s, W1, b1s, W2s, b2s, lane);
    slab_load(X, kv, lane);
    #pragma unroll
    for (int i = 0; i < 32; ++i) acc[i] += 2.f * kv[i];

    // k3 = f(t + dt/2, xc + dt/2 * k2)
    #pragma unroll
    for (int i = 0; i < 32; ++i) kv[i] = xc[i] + 0.5f * dt * kv[i];
    slab_store(X, kv, t + 0.5f * dt, lane);
    f_eval(X, h0, h1, W0s, b0s, W1, b1s, W2s, b2s, lane);
    slab_load(X, kv, lane);
    #pragma unroll
    for (int i = 0; i < 32; ++i) acc[i] += 2.f * kv[i];

    // k4 = f(t + dt, xc + dt * k3)
    #pragma unroll
    for (int i = 0; i < 32; ++i) kv[i] = xc[i] + dt * kv[i];
    slab_store(X, kv, t + dt, lane);
    f_eval(X, h0, h1, W0s, b0s, W1, b1s, W2s, b2s, lane);
    slab_load(X, kv, lane);
    #pragma unroll
    for (int i = 0; i < 32; ++i) acc[i] += kv[i];

    #pragma unroll
    for (int i = 0; i < 32; ++i) xc[i] += dt * acc[i] * (1.0f / 6.0f);
  }

  // ---- write outputs: x_out [65536*64] then log_det [65536], concatenated flat
  #pragma unroll
  for (int i = 0; i < 32; ++i) out[slab + lane + 32 * i] = xc[i];
  if (lane < 16) out[(size_t)BTOT * DD + rowbase + lane] = ld;
}

extern "C" void kernel_launch(void* const* d_in, const int* in_sizes, int n_in,
                              void* d_out, int out_size, void* d_ws, size_t ws_size,
                              hipStream_t stream) {
  const float* x  = (const float*)d_in[0];
  const float* ep = (const float*)d_in[1];
  const float* W0 = (const float*)d_in[2];
  const float* b0 = (const float*)d_in[3];
  const float* W1 = (const float*)d_in[4];
  const float* b1 = (const float*)d_in[5];
  const float* W2 = (const float*)d_in[6];
  const float* b2 = (const float*)d_in[7];
  float* out = (float*)d_out;

  dim3 grid(BTOT / BT);     // 1024 blocks
  dim3 blk(NW * 32);        // 4 wave32 waves
  hipLaunchKernelGGL(ffjord_rk4_wmma, grid, blk, 0, stream,
                     x, ep, W0, b0, W1, b1, W2, b2, out);
}